// MaskEncoder_15693810499988
// MI455X (gfx1250) — compile-verified
//
#include <hip/hip_runtime.h>
#include <hip/hip_bf16.h>
#include <cstdint>
#include <cstddef>

// ---------------- problem constants ----------------
namespace {
constexpr int NN    = 100000;        // nodes
constexpr int NE    = 1600000;       // edges
constexpr int FIN   = 128;
constexpr int FH    = 64;
constexpr int FO    = 16;
constexpr int EALL  = NE + NN;       // edges + self loops
constexpr int KSEL  = 800000;        // k_homo == k_hetero == E/2
constexpr int NPADN = 1 << 21;       // 2097152, bitonic pad
constexpr int PADN  = NPADN - NE;    // 497152 pad entries (smallest keys)
constexpr int CHUNK = 8192;          // LDS bitonic chunk (64 KB of LDS keys+payloads)
constexpr int NT    = 1024;          // threads per bitonic block (32 waves)
constexpr int ELTS  = CHUNK / NT;    // elements per thread per array (8)
constexpr int PAIRS = CHUNK / 2 / NT;// compare-exchanges per thread per substage (4)
}

typedef float v2f __attribute__((ext_vector_type(2)));
typedef float v8f __attribute__((ext_vector_type(8)));
typedef unsigned tdm_u4 __attribute__((ext_vector_type(4)));
typedef int      tdm_i8 __attribute__((ext_vector_type(8)));
typedef int      tdm_i4 __attribute__((ext_vector_type(4)));

// ---------------- helpers ----------------
__device__ __forceinline__ unsigned f2ord(float f) {
  unsigned u = __float_as_uint(f);
  return (u & 0x80000000u) ? ~u : (u | 0x80000000u);   // monotonic float->uint
}
__device__ __forceinline__ float ord2f(unsigned e) {
  return (e & 0x80000000u) ? __uint_as_float(e ^ 0x80000000u) : __uint_as_float(~e);
}
__device__ __forceinline__ float lrelu(float v, float s) { return v > 0.0f ? v : v * s; }

__device__ __forceinline__ void edge_sd(const int* __restrict__ ei, int idx, int& s, int& d) {
  if (idx < NE) { s = ei[idx]; d = ei[NE + idx]; }
  else          { s = d = idx - NE; }                  // appended self loops
}

// ---------------- WMMA f32 GEMM: C[N x NCOL] = A[N x K] @ W[K x NCOL] ----------------
// One wave computes a 16x16 f32 tile with V_WMMA_F32_16X16X4_F32 (K-step 4).
template <int K, int NCOL>
__global__ void k_gemm_wmma(const float* __restrict__ A, const float* __restrict__ W,
                            float* __restrict__ C) {
  const int  lane = threadIdx.x & 31;
  const int  wave = threadIdx.x >> 5;
  const int  col0 = wave * 16;
  const int  m    = lane & 15;
  const int  half = lane >> 4;
  const long row0 = (long)blockIdx.x * 16;
  const float* __restrict__ arow = A + (row0 + m) * K;

  v8f acc = {};
  for (int k0 = 0; k0 < K; k0 += 4) {
    const int kk = k0 + half * 2;
    v2f a, b;
    a[0] = arow[kk];
    a[1] = arow[kk + 1];
    b[0] = W[(size_t)kk * NCOL + col0 + m];
    b[1] = W[(size_t)(kk + 1) * NCOL + col0 + m];
    acc = __builtin_amdgcn_wmma_f32_16x16x4_f32(false, a, false, b, (short)0, acc,
                                                false, false);
  }
  const int  col = col0 + m;
  const long rb  = row0 + half * 8;
#pragma unroll
  for (int r = 0; r < 8; ++r) C[(rb + r) * NCOL + col] = acc[r];
}

// ---------------- small utility kernels ----------------
__global__ void k_fill_u32(unsigned* __restrict__ p, unsigned v, long n) {
  long i = (long)blockIdx.x * blockDim.x + threadIdx.x;
  if (i < n) p[i] = v;
}

// per-node attention projections: es = h . a_src, ed = h . a_dst (fully unrolled)
template <int F>
__global__ void k_attvec(const float* __restrict__ h, const float* __restrict__ asrc,
                         const float* __restrict__ adst, float* __restrict__ es,
                         float* __restrict__ ed) {
  int n = blockIdx.x * blockDim.x + threadIdx.x;
  if (n >= NN) return;
  const float4* hp = (const float4*)(h + (size_t)n * F);
  float s = 0.0f, d = 0.0f;
#pragma unroll
  for (int c = 0; c < F / 4; ++c) {
    float4 v = hp[c];
    s += v.x * asrc[4 * c] + v.y * asrc[4 * c + 1] + v.z * asrc[4 * c + 2] +
         v.w * asrc[4 * c + 3];
    d += v.x * adst[4 * c] + v.y * adst[4 * c + 1] + v.z * adst[4 * c + 2] +
         v.w * adst[4 * c + 3];
  }
  es[n] = s; ed[n] = d;
}

// segment max over destinations (order-encoded uint atomicMax)
__global__ void k_edge_max(const int* __restrict__ ei, const float* __restrict__ es,
                           const float* __restrict__ ed, unsigned* __restrict__ mseg) {
  int idx = blockIdx.x * blockDim.x + threadIdx.x;
  if (idx >= EALL) return;
  int s, d; edge_sd(ei, idx, s, d);
  float e = lrelu(es[s] + ed[d], 0.2f);
  atomicMax(&mseg[d], f2ord(e));
}

// ee = exp(e - m[dst]); segment sum
__global__ void k_edge_exp(const int* __restrict__ ei, const float* __restrict__ es,
                           const float* __restrict__ ed, const unsigned* __restrict__ mseg,
                           float* __restrict__ sseg, float* __restrict__ ee) {
  int idx = blockIdx.x * blockDim.x + threadIdx.x;
  if (idx >= EALL) return;
  int s, d; edge_sd(ei, idx, s, d);
  float e  = lrelu(es[s] + ed[d], 0.2f);
  float x  = __expf(e - ord2f(mseg[d]));
  ee[idx]  = x;
  atomicAdd(&sseg[d], x);
}

// out[dst] += (ee/(s[dst]+eps)) * h[src]   (float4 chunks; alpha fused, sseg L2-resident)
__global__ void k_scatter(const float* __restrict__ h, const float* __restrict__ ee,
                          const float* __restrict__ sseg, const int* __restrict__ ei,
                          float* __restrict__ out, int F4) {
  long t = (long)blockIdx.x * blockDim.x + threadIdx.x;
  long total = (long)EALL * F4;
  if (t >= total) return;
  int idx = (int)(t / F4);
  int c   = (int)(t % F4);
  int s, d; edge_sd(ei, idx, s, d);
  float a = ee[idx] / (sseg[d] + 1e-16f);
  float4 hv = ((const float4*)(h + (size_t)s * F4 * 4))[c];
  float* op = out + (size_t)d * F4 * 4 + (size_t)c * 4;
  atomicAdd(op + 0, hv.x * a);
  atomicAdd(op + 1, hv.y * a);
  atomicAdd(op + 2, hv.z * a);
  atomicAdd(op + 3, hv.w * a);
}

// x = lrelu(x + b[f], slope); optional mirror copy into dst2 (x2 -> d_out)
__global__ void k_bias_act(float* __restrict__ x, const float* __restrict__ b, int F,
                           float slope, float* __restrict__ dst2, long n) {
  long i = (long)blockIdx.x * blockDim.x + threadIdx.x;
  if (i >= n) return;
  float v = lrelu(x[i] + b[i % F], slope);
  x[i] = v;
  if (dst2) dst2[i] = v;
}

// per-edge homophily score -> (ordered key, edge id), pad with smallest keys
__global__ void k_value(const float* __restrict__ x2, const int* __restrict__ ei,
                        unsigned* __restrict__ key, unsigned* __restrict__ pay) {
  int t = blockIdx.x * blockDim.x + threadIdx.x;
  if (t >= NPADN) return;
  if (t < NE) {
    int s = ei[t], d = ei[NE + t];
    const float4* ps = (const float4*)(x2 + (size_t)s * FO);
    const float4* pd = (const float4*)(x2 + (size_t)d * FO);
    float v = 0.0f;
#pragma unroll
    for (int c = 0; c < FO / 4; ++c) {
      float4 a = ps[c], b = pd[c];
      v += a.x * b.x + a.y * b.y + a.z * b.z + a.w * b.w;
    }
    key[t] = f2ord(v);
    pay[t] = (unsigned)t;
  } else {
    key[t] = 0u;           // strictly below any real encoded float
    pay[t] = 0u;
  }
}

// ---------------- TDM chunk staging (gfx1250 Tensor Data Mover) ----------------
// 1-D D#: data_size=4B, tile_dim0=CHUNK, tensor_dim0=CHUNK, stride=CHUNK.
// Group0: [1:0]=count=1, [63:32]=lds_addr, [120:64]=global_addr, [127:126]=type 2.
// LDS byte offset = low 32 bits of the flat LDS pointer (flat LDS aperture rule).
#if defined(__has_builtin)
#if __has_builtin(__builtin_amdgcn_tensor_load_to_lds) && \
    __has_builtin(__builtin_amdgcn_s_wait_tensorcnt)
#define USE_TDM 1
#endif
#endif

#ifdef USE_TDM
__device__ __forceinline__ void tdm_load_chunk(const unsigned* __restrict__ gsrc,
                                               unsigned* lds_dst) {
  unsigned long long ga = (unsigned long long)(uintptr_t)gsrc;
  unsigned la = (unsigned)(uintptr_t)lds_dst;      // LDS byte offset
  tdm_u4 g0;
  g0[0] = 1u;                                      // count=1 (user descriptor)
  g0[1] = la;                                      // lds_addr
  g0[2] = (unsigned)ga;                            // global_addr[31:0]
  g0[3] = (unsigned)((ga >> 32) & 0x01FFFFFFu) | 0x80000000u;  // addr[56:32] | type=2
  tdm_i8 g1;
  g1[0] = 0x20000;                                 // data_size=2 (4 bytes), no flags
  g1[1] = (CHUNK & 0xFFFF) << 16;                  // tensor_dim0[15:0] @ bit48
  g1[2] = ((CHUNK >> 16) & 0xFFFF) | (1 << 16);    // tensor_dim0[31:16], tensor_dim1=1
  g1[3] = (CHUNK & 0xFFFF) << 16;                  // tile_dim0 = CHUNK @ bit112
  g1[4] = 0;                                       // tile_dim1/2 unused
  g1[5] = CHUNK;                                   // tensor_dim0_stride[31:0]
  g1[6] = 0;
  g1[7] = 0;
  tdm_i4 z4 = {};
  tdm_i8 z8 = {};
  // 6-arg form (clang-23 / therock-10.0 HIP headers)
  __builtin_amdgcn_tensor_load_to_lds(g0, g1, z4, z4, z8, 0);
}
#endif

// Stage a CHUNK of keys+payloads into LDS: one TDM DMA per array (TDM is per-wave
// and ignores EXEC, so wave 0 issues exactly one load per descriptor), then
// s_wait_tensorcnt + workgroup barrier. Fallback: per-lane staged loads.
__device__ __forceinline__ void chunk_load(const unsigned* __restrict__ key,
                                           const unsigned* __restrict__ pay,
                                           unsigned* sk, unsigned* sp, int base) {
#ifdef USE_TDM
  if (threadIdx.x == 0) {
    tdm_load_chunk(key + base, sk);
    tdm_load_chunk(pay + base, sp);
  }
  __builtin_amdgcn_s_wait_tensorcnt(0);
#else
#pragma unroll
  for (int u = 0; u < ELTS; ++u) {
    int t = threadIdx.x + u * NT;
    sk[t] = key[base + t];
    sp[t] = pay[base + t];
  }
#endif
  __syncthreads();
}

__device__ __forceinline__ void chunk_store(unsigned* __restrict__ key,
                                            unsigned* __restrict__ pay,
                                            const unsigned* sk, const unsigned* sp,
                                            int base) {
#pragma unroll
  for (int u = 0; u < ELTS; ++u) {
    int t = threadIdx.x + u * NT;
    key[base + t] = sk[t];
    pay[base + t] = sp[t];
  }
}

__device__ __forceinline__ void lds_substage(unsigned* sk, unsigned* sp, int base,
                                             int j, int k) {
#pragma unroll
  for (int u = 0; u < PAIRS; ++u) {
    int tt = threadIdx.x + u * NT;
    int i = ((tt & ~(j - 1)) << 1) | (tt & (j - 1));
    int p = i | j;
    bool asc = (((base + i) & k) == 0);
    unsigned ki = sk[i], kj = sk[p];
    if ((ki > kj) == asc) {
      sk[i] = kj; sk[p] = ki;
      unsigned q = sp[i]; sp[i] = sp[p]; sp[p] = q;
    }
  }
  __syncthreads();
}

// ---------------- hybrid bitonic sort ----------------
// Global compare-exchange pass (only needed for strides j >= CHUNK)
__global__ void k_bitonic_global(unsigned* __restrict__ key, unsigned* __restrict__ pay,
                                 int j, int k) {
  int i = blockIdx.x * blockDim.x + threadIdx.x;
  int ixj = i ^ j;
  if (ixj > i) {
    bool asc = ((i & k) == 0);
    unsigned ki = key[i], kj = key[ixj];
    if ((ki > kj) == asc) {
      key[i] = kj; key[ixj] = ki;
      unsigned pi = pay[i]; pay[i] = pay[ixj]; pay[ixj] = pi;
    }
  }
}

// All stages k=2..CHUNK entirely in LDS (one staging in, one write-back)
__global__ __launch_bounds__(NT) void k_bitonic_lds_full(unsigned* __restrict__ key,
                                                         unsigned* __restrict__ pay) {
  __shared__ unsigned sk[CHUNK];
  __shared__ unsigned sp[CHUNK];
  const int base = blockIdx.x * CHUNK;
  chunk_load(key, pay, sk, sp, base);
  for (int k = 2; k <= CHUNK; k <<= 1)
    for (int j = k >> 1; j > 0; j >>= 1)
      lds_substage(sk, sp, base, j, k);
  chunk_store(key, pay, sk, sp, base);
}

// Tail sub-stages j = CHUNK/2 .. 1 of stage k, entirely in LDS
__global__ __launch_bounds__(NT) void k_bitonic_lds_tail(unsigned* __restrict__ key,
                                                         unsigned* __restrict__ pay,
                                                         int k) {
  __shared__ unsigned sk[CHUNK];
  __shared__ unsigned sp[CHUNK];
  const int base = blockIdx.x * CHUNK;
  chunk_load(key, pay, sk, sp, base);
  for (int j = CHUNK / 2; j > 0; j >>= 1)
    lds_substage(sk, sp, base, j, k);
  chunk_store(key, pay, sk, sp, base);
}

// emit edge_index columns: homo = largest KSEL (descending), hetero = smallest (ascending)
__global__ void k_emit(const unsigned* __restrict__ pay, const int* __restrict__ ei,
                       float* __restrict__ out) {
  int j = blockIdx.x * blockDim.x + threadIdx.x;
  if (j >= KSEL) return;
  int eh = (int)pay[NPADN - 1 - j];       // descending by value
  out[j]            = (float)ei[eh];
  out[KSEL + j]     = (float)ei[NE + eh];
  int el = (int)pay[PADN + j];            // ascending by value (skip pads)
  out[2 * KSEL + j] = (float)ei[el];
  out[3 * KSEL + j] = (float)ei[NE + el];
}

// ---------------- host ----------------
extern "C" void kernel_launch(void* const* d_in, const int* in_sizes, int n_in,
                              void* d_out, int out_size, void* d_ws, size_t ws_size,
                              hipStream_t stream) {
  (void)in_sizes; (void)n_in; (void)out_size; (void)ws_size;
  const float* x   = (const float*)d_in[0];
  const int*   ei  = (const int*)d_in[1];
  const float* W1  = (const float*)d_in[2];
  const float* as1 = (const float*)d_in[3];
  const float* ad1 = (const float*)d_in[4];
  const float* b1  = (const float*)d_in[5];
  const float* W2  = (const float*)d_in[6];
  const float* as2 = (const float*)d_in[7];
  const float* ad2 = (const float*)d_in[8];
  const float* b2  = (const float*)d_in[9];
  float* out = (float*)d_out;

  // workspace carve-out (~89 MB)
  char* w = (char*)d_ws;
  auto carve = [&](size_t nbytes) {
    char* p = w; w += (nbytes + 255) & ~(size_t)255; return p;
  };
  float*    h1   = (float*)carve((size_t)NN * FH * 4);
  float*    x1   = (float*)carve((size_t)NN * FH * 4);   // agg accumulator -> x1 in place
  float*    h2   = (float*)carve((size_t)NN * FO * 4);
  float*    x2   = (float*)carve((size_t)NN * FO * 4);   // agg accumulator -> x2 in place
  float*    es   = (float*)carve((size_t)NN * 4);        // reused by both layers
  float*    ed   = (float*)carve((size_t)NN * 4);
  unsigned* mseg = (unsigned*)carve((size_t)NN * 4);
  float*    sseg = (float*)carve((size_t)NN * 4);
  float*    ee   = (float*)carve((size_t)EALL * 4);
  unsigned* keys = (unsigned*)carve((size_t)NPADN * 4);
  unsigned* pays = (unsigned*)carve((size_t)NPADN * 4);

  const int B = 256;
  auto nb = [](long n, int b) { return (int)((n + b - 1) / b); };

  // ----- layer 1: h1 = x @ W1 (WMMA f32), attention softmax, scatter, act -----
  k_fill_u32<<<nb((long)NN * FH, B), B, 0, stream>>>((unsigned*)x1, 0u, (long)NN * FH);
  k_fill_u32<<<nb(NN, B), B, 0, stream>>>(mseg, 0u, NN);
  k_fill_u32<<<nb(NN, B), B, 0, stream>>>((unsigned*)sseg, 0u, NN);
  k_gemm_wmma<FIN, FH><<<NN / 16, 128, 0, stream>>>(x, W1, h1);     // 4 waves/block
  k_attvec<FH><<<nb(NN, B), B, 0, stream>>>(h1, as1, ad1, es, ed);
  k_edge_max<<<nb(EALL, B), B, 0, stream>>>(ei, es, ed, mseg);
  k_edge_exp<<<nb(EALL, B), B, 0, stream>>>(ei, es, ed, mseg, sseg, ee);
  k_scatter<<<nb((long)EALL * (FH / 4), B), B, 0, stream>>>(h1, ee, sseg, ei, x1, FH / 4);
  k_bias_act<<<nb((long)NN * FH, B), B, 0, stream>>>(x1, b1, FH, 0.01f, nullptr,
                                                     (long)NN * FH);

  // ----- layer 2: h2 = x1 @ W2 (WMMA f32), attention softmax, scatter, +b2 -----
  k_fill_u32<<<nb((long)NN * FO, B), B, 0, stream>>>((unsigned*)x2, 0u, (long)NN * FO);
  k_fill_u32<<<nb(NN, B), B, 0, stream>>>(mseg, 0u, NN);
  k_fill_u32<<<nb(NN, B), B, 0, stream>>>((unsigned*)sseg, 0u, NN);
  k_gemm_wmma<FH, FO><<<NN / 16, 32, 0, stream>>>(x1, W2, h2);      // 1 wave/block
  k_attvec<FO><<<nb(NN, B), B, 0, stream>>>(h2, as2, ad2, es, ed);
  k_edge_max<<<nb(EALL, B), B, 0, stream>>>(ei, es, ed, mseg);
  k_edge_exp<<<nb(EALL, B), B, 0, stream>>>(ei, es, ed, mseg, sseg, ee);
  k_scatter<<<nb((long)EALL * (FO / 4), B), B, 0, stream>>>(h2, ee, sseg, ei, x2, FO / 4);
  // slope 1.0 => identity activation; mirror x2 into d_out tail
  k_bias_act<<<nb((long)NN * FO, B), B, 0, stream>>>(x2, b2, FO, 1.0f,
                                                     out + 4L * KSEL, (long)NN * FO);

  // ----- per-edge score, hybrid TDM/LDS/global bitonic sort, emit top-k halves -----
  k_value<<<NPADN / B, B, 0, stream>>>(x2, ei, keys, pays);
  // stages k=2..CHUNK fully in LDS (TDM-staged)
  k_bitonic_lds_full<<<NPADN / CHUNK, NT, 0, stream>>>(keys, pays);
  // remaining stages: global passes only while stride >= CHUNK, then LDS tail
  for (int k = CHUNK * 2; k <= NPADN; k <<= 1) {
    for (int j = k >> 1; j >= CHUNK; j >>= 1)
      k_bitonic_global<<<NPADN / B, B, 0, stream>>>(keys, pays, j, k);
    k_bitonic_lds_tail<<<NPADN / CHUNK, NT, 0, stream>>>(keys, pays, k);
  }
  k_emit<<<nb(KSEL, B), B, 0, stream>>>(pays, ei, out);
}